// GCN_60344290509164
// MI455X (gfx1250) — compile-verified
//
#include <hip/hip_runtime.h>
#include <hip/hip_bf16.h>
#include <math.h>

// ---------------------------------------------------------------------------
// GCN pipeline for MI455X (gfx1250), all fp32, matrix ops via
// V_WMMA_F32_16X16X4_F32 (wave32 WMMA).
//
// Sizes (from reference): N=64000 nodes, F0=1000, F1=32, F2=16, E=3.2M edges,
// B=64 graphs, MLP: 16000->64->1.
// ---------------------------------------------------------------------------

typedef __attribute__((ext_vector_type(2))) float v2f;
typedef __attribute__((ext_vector_type(8))) float v8f;

#define N_NODES 64000
#define F0 1000
#define F1 32
#define F2 16
#define N_EDGES 3200000
#define N_GRAPHS 64
#define MLP_K 16000
#define MLP_H 64

// ---------------- degree / normalization --------------------------------
__global__ __launch_bounds__(256) void k_init_deg(float* deg) {
  int i = blockIdx.x * 256 + threadIdx.x;
  if (i < N_NODES) deg[i] = 1.0f;  // self-loop weight
}

__global__ __launch_bounds__(256) void k_scatter_deg(const long long* __restrict__ col,
                                                     const float* __restrict__ ew,
                                                     float* __restrict__ deg) {
  int e = blockIdx.x * 256 + threadIdx.x;
  if (e < N_EDGES) atomicAdd(&deg[(int)col[e]], ew[e]);
}

__global__ __launch_bounds__(256) void k_rsqrt(float* deg) {
  int i = blockIdx.x * 256 + threadIdx.x;
  if (i < N_NODES) {
    float d = deg[i];
    deg[i] = (d > 0.0f) ? rsqrtf(d) : 0.0f;  // deg -> dinv in place
  }
}

// ---------------- GEMM1: xl1[64000,32] = x @ W1  (K=1000) ----------------
// wave = 64 rows x 32 cols (4 m-tiles x 2 n-tiles), 1000 waves, 125 blocks.
__global__ __launch_bounds__(256) void k_gemm1(const float* __restrict__ x,
                                               const float* __restrict__ W1,
                                               float* __restrict__ xl) {
  const int wave = threadIdx.x >> 5;
  const int lane = threadIdx.x & 31;
  const int gw   = blockIdx.x * 8 + wave;      // 0..999
  const int row0 = gw * 64;
  const int ml   = lane & 15;
  const int kk   = (lane >> 4) << 1;           // 0 or 2

  v8f acc[4][2] = {};
  const float* xr0 = x + (size_t)(row0 + ml) * F0 + kk;

  for (int k0 = 0; k0 < F0; k0 += 4) {
    // prefetch the streamed A operand ahead (global_prefetch_b8)
    if ((k0 & 63) == 0) __builtin_prefetch(xr0 + k0 + 128, 0, 1);

    v2f a[4];
#pragma unroll
    for (int mt = 0; mt < 4; ++mt)
      a[mt] = *(const v2f*)(xr0 + (size_t)mt * 16 * F0 + k0);

    const float* wp = W1 + (size_t)(k0 + kk) * F1;
    v2f b0, b1;
    b0.x = wp[ml];        b0.y = wp[F1 + ml];        // n-tile 0
    b1.x = wp[16 + ml];   b1.y = wp[F1 + 16 + ml];   // n-tile 1

#pragma unroll
    for (int mt = 0; mt < 4; ++mt) {
      acc[mt][0] = __builtin_amdgcn_wmma_f32_16x16x4_f32(
          false, a[mt], false, b0, (short)0, acc[mt][0], false, false);
      acc[mt][1] = __builtin_amdgcn_wmma_f32_16x16x4_f32(
          false, a[mt], false, b1, (short)0, acc[mt][1], false, false);
    }
  }

  const int rj = (lane >> 4) * 8;
#pragma unroll
  for (int mt = 0; mt < 4; ++mt)
#pragma unroll
    for (int nt = 0; nt < 2; ++nt)
#pragma unroll
      for (int j = 0; j < 8; ++j)
        xl[(size_t)(row0 + mt * 16 + rj + j) * F1 + nt * 16 + ml] = acc[mt][nt][j];
}

// ---------------- self-loop init:  out = dinv^2 * xl ---------------------
template <int F>
__global__ __launch_bounds__(256) void k_init_out(const float* __restrict__ xl,
                                                  const float* __restrict__ dinv,
                                                  float* __restrict__ out) {
  int i = blockIdx.x * 256 + threadIdx.x;
  if (i < N_NODES * F) {
    float d = dinv[i / F];
    out[i] = d * d * xl[i];
  }
}

// ---------------- edge scatter, 32 features: one wave per edge -----------
__global__ __launch_bounds__(256) void k_scatter32(const long long* __restrict__ row,
                                                   const long long* __restrict__ col,
                                                   const float* __restrict__ ew,
                                                   const float* __restrict__ dinv,
                                                   const float* __restrict__ xl,
                                                   float* __restrict__ out) {
  const int wave = threadIdx.x >> 5;
  const int lane = threadIdx.x & 31;
  const int e = blockIdx.x * 8 + wave;
  if (e >= N_EDGES) return;
  const int r = (int)row[e];
  const int c = (int)col[e];
  const float norm = dinv[r] * ew[e] * dinv[c];
  atomicAdd(&out[(size_t)c * 32 + lane], norm * xl[(size_t)r * 32 + lane]);
}

// ---------------- edge scatter, 16 features: two edges per wave ----------
__global__ __launch_bounds__(256) void k_scatter16(const long long* __restrict__ row,
                                                   const long long* __restrict__ col,
                                                   const float* __restrict__ ew,
                                                   const float* __restrict__ dinv,
                                                   const float* __restrict__ xl,
                                                   float* __restrict__ out) {
  const int wave = threadIdx.x >> 5;
  const int lane = threadIdx.x & 31;
  const int e = blockIdx.x * 16 + wave * 2 + (lane >> 4);
  const int f = lane & 15;
  if (e >= N_EDGES) return;
  const int r = (int)row[e];
  const int c = (int)col[e];
  const float norm = dinv[r] * ew[e] * dinv[c];
  atomicAdd(&out[(size_t)c * 16 + f], norm * xl[(size_t)r * 16 + f]);
}

// ---------------- bias + relu (in place) ---------------------------------
template <int F>
__global__ __launch_bounds__(256) void k_finalize(float* __restrict__ h,
                                                  const float* __restrict__ b) {
  int i = blockIdx.x * 256 + threadIdx.x;
  if (i < N_NODES * F) {
    float v = h[i] + b[i & (F - 1)];
    h[i] = v > 0.0f ? v : 0.0f;
  }
}

// ---------------- GEMM2: xl2[64000,16] = h1 @ W2  (K=32) -----------------
// wave = 32 rows x 16 cols (2 m-tiles), 2000 waves, 250 blocks.
__global__ __launch_bounds__(256) void k_gemm2(const float* __restrict__ h1,
                                               const float* __restrict__ W2,
                                               float* __restrict__ xl2) {
  const int wave = threadIdx.x >> 5;
  const int lane = threadIdx.x & 31;
  const int gw   = blockIdx.x * 8 + wave;      // 0..1999
  const int row0 = gw * 32;
  const int ml   = lane & 15;
  const int kk   = (lane >> 4) << 1;

  v8f acc[2] = {};
#pragma unroll
  for (int k0 = 0; k0 < F1; k0 += 4) {
    v2f a[2];
#pragma unroll
    for (int mt = 0; mt < 2; ++mt)
      a[mt] = *(const v2f*)(h1 + (size_t)(row0 + mt * 16 + ml) * F1 + k0 + kk);
    const float* wp = W2 + (size_t)(k0 + kk) * F2;
    v2f b;
    b.x = wp[ml]; b.y = wp[F2 + ml];
#pragma unroll
    for (int mt = 0; mt < 2; ++mt)
      acc[mt] = __builtin_amdgcn_wmma_f32_16x16x4_f32(
          false, a[mt], false, b, (short)0, acc[mt], false, false);
  }

  const int rj = (lane >> 4) * 8;
#pragma unroll
  for (int mt = 0; mt < 2; ++mt)
#pragma unroll
    for (int j = 0; j < 8; ++j)
      xl2[(size_t)(row0 + mt * 16 + rj + j) * F2 + ml] = acc[mt][j];
}

// ---------------- MLP1: t[64,64] = relu(h2mat[64,16000] @ Wf1 + bf1) -----
// 16 waves (4x4 tiles of 16x16), K=16000.
__global__ __launch_bounds__(256) void k_mlp1(const float* __restrict__ h2,
                                              const float* __restrict__ Wf1,
                                              const float* __restrict__ bf1,
                                              float* __restrict__ t) {
  const int wave = threadIdx.x >> 5;
  const int lane = threadIdx.x & 31;
  const int gw   = blockIdx.x * 8 + wave;  // 0..15
  const int mt   = gw >> 2;
  const int nt   = gw & 3;
  const int ml   = lane & 15;
  const int kk   = (lane >> 4) << 1;

  v8f acc = {};
  const float* ar = h2 + (size_t)(mt * 16 + ml) * MLP_K + kk;
  for (int k0 = 0; k0 < MLP_K; k0 += 4) {
    if ((k0 & 63) == 0) __builtin_prefetch(ar + k0 + 128, 0, 1);
    v2f a = *(const v2f*)(ar + k0);
    const float* wp = Wf1 + (size_t)(k0 + kk) * MLP_H + nt * 16;
    v2f b;
    b.x = wp[ml]; b.y = wp[MLP_H + ml];
    acc = __builtin_amdgcn_wmma_f32_16x16x4_f32(
        false, a, false, b, (short)0, acc, false, false);
  }

  const int rj = (lane >> 4) * 8;
#pragma unroll
  for (int j = 0; j < 8; ++j) {
    int r = mt * 16 + rj + j;
    int c = nt * 16 + ml;
    float v = acc[j] + bf1[c];
    t[r * MLP_H + c] = v > 0.0f ? v : 0.0f;
  }
}

// ---------------- MLP2: out[64] = sigmoid(t @ Wf2 + bf2) -----------------
__global__ __launch_bounds__(64) void k_mlp2(const float* __restrict__ t,
                                             const float* __restrict__ Wf2,
                                             const float* __restrict__ bf2,
                                             float* __restrict__ out) {
  int g = threadIdx.x;
  if (g < N_GRAPHS) {
    float s = bf2[0];
#pragma unroll 8
    for (int j = 0; j < MLP_H; ++j) s += t[g * MLP_H + j] * Wf2[j];
    out[g] = 1.0f / (1.0f + expf(-s));
  }
}

// ---------------------------------------------------------------------------
extern "C" void kernel_launch(void* const* d_in, const int* in_sizes, int n_in,
                              void* d_out, int out_size, void* d_ws, size_t ws_size,
                              hipStream_t stream) {
  const float*     x   = (const float*)d_in[0];
  const long long* ei  = (const long long*)d_in[1];   // [2, E] int64
  const float*     ew  = (const float*)d_in[2];
  const float*     W1  = (const float*)d_in[3];
  const float*     b1  = (const float*)d_in[4];
  const float*     W2  = (const float*)d_in[5];
  const float*     b2  = (const float*)d_in[6];
  const float*     Wf1 = (const float*)d_in[7];
  const float*     bf1 = (const float*)d_in[8];
  const float*     Wf2 = (const float*)d_in[9];
  const float*     bf2 = (const float*)d_in[10];
  float* out = (float*)d_out;

  const long long* row = ei;
  const long long* col = ei + N_EDGES;

  // workspace layout (floats), regions reused across layers: ~16.7 MB total
  float* ws   = (float*)d_ws;
  float* dinv = ws;                                   // 64000
  float* xl1  = ws + N_NODES;                         // 2,048,000 (layer1 xW)
  float* h1   = xl1 + (size_t)N_NODES * F1;           // 2,048,000 (out1 -> h1)
  float* xl2  = xl1;                                  // 1,024,000 (reuse xl1 region)
  float* h2   = xl1 + (size_t)N_NODES * F2;           // 1,024,000 (reuse xl1 region)
  float* t    = h1 + (size_t)N_NODES * F1;            // 4096

  // --- normalization ---
  k_init_deg<<<(N_NODES + 255) / 256, 256, 0, stream>>>(dinv);
  k_scatter_deg<<<(N_EDGES + 255) / 256, 256, 0, stream>>>(col, ew, dinv);
  k_rsqrt<<<(N_NODES + 255) / 256, 256, 0, stream>>>(dinv);

  // --- layer 1 ---
  k_gemm1<<<N_NODES / 64 / 8, 256, 0, stream>>>(x, W1, xl1);
  k_init_out<F1><<<(N_NODES * F1 + 255) / 256, 256, 0, stream>>>(xl1, dinv, h1);
  k_scatter32<<<(N_EDGES + 7) / 8, 256, 0, stream>>>(row, col, ew, dinv, xl1, h1);
  k_finalize<F1><<<(N_NODES * F1 + 255) / 256, 256, 0, stream>>>(h1, b1);

  // --- layer 2 ---
  k_gemm2<<<N_NODES / 32 / 8, 256, 0, stream>>>(h1, W2, xl2);
  k_init_out<F2><<<(N_NODES * F2 + 255) / 256, 256, 0, stream>>>(xl2, dinv, h2);
  k_scatter16<<<(N_EDGES + 15) / 16, 256, 0, stream>>>(row, col, ew, dinv, xl2, h2);
  k_finalize<F2><<<(N_NODES * F2 + 255) / 256, 256, 0, stream>>>(h2, b2);

  // --- MLP head ---
  k_mlp1<<<2, 256, 0, stream>>>(h2, Wf1, bf1, t);
  k_mlp2<<<1, 64, 0, stream>>>(t, Wf2, bf2, out);
}